// CMCD_55602646614308
// MI455X (gfx1250) — compile-verified
//
#include <hip/hip_runtime.h>
#include <math.h>

typedef __attribute__((ext_vector_type(2))) float v2f;
typedef __attribute__((ext_vector_type(8))) float v8f;
typedef __attribute__((ext_vector_type(4))) int   i32x4;

#define Bn   512          // particles
#define Dd   64           // dim
#define Cc   512          // channels
#define NB   8            // steps
#define LOG2PI 1.8378770664093453f
#define ASTR 36           // padded A-tile row stride (dwords): 36*r mod 64 distinct for r<16

#ifndef __has_builtin
#define __has_builtin(x) 0
#endif

#if defined(__HIP_DEVICE_COMPILE__) && __has_builtin(__builtin_amdgcn_global_load_async_to_lds_b128)
#define HAVE_ASYNC_LDS 1
#else
#define HAVE_ASYNC_LDS 0
#endif

__device__ __forceinline__ void wait_async_zero() {
#if HAVE_ASYNC_LDS
#if __has_builtin(__builtin_amdgcn_s_wait_asynccnt)
    __builtin_amdgcn_s_wait_asynccnt(0);
#else
    asm volatile("s_wait_asynccnt 0x0" ::: "memory");
#endif
#endif
}

__device__ __forceinline__ float gelu_f(float x) {
    // jax.nn.gelu default (approximate=True, tanh form)
    float x3 = x * x * x;
    return 0.5f * x * (1.0f + tanhf(0.7978845608028654f * (x + 0.044715f * x3)));
}

// ---------------------------------------------------------------------------
// WMMA GEMM: Out = epilogue( A[512 x K] @ B[K x N] )
// Block = 128 threads (4 waves). Block tile = 64(M) x 64(N).
// B tile staged in LDS, k-pair interleaved: lb[(k>>1)*128 + col*2 + (k&1)]
//   -> each WMMA B fragment is a single aligned ds_load_b64, no shuffles.
// A tile staged via async global->LDS DMA (padded stride, bank-conflict-free),
//   falling back to direct global b64 loads when the builtin is unavailable.
// EPI 0: gelu(val + bias[col] + te[col])          -> Out[512 x N], N=512
// EPI 1: xn = x + dt*(glp - (val+bias)) + s2*nz   -> Out[512 x 64]
// EPI 2: d2 = rn[row] + rn[col] - 2*val           -> Out[512 x 512] (BT: B=X^T)
// EPI 3: xo = xn + coef*(rowsum[row]*xold - val)  -> Out[512 x 64]
// ---------------------------------------------------------------------------
template <int EPI, bool BT>
__global__ __launch_bounds__(128) void gemm_k(
    const float* __restrict__ A, const float* __restrict__ Bm,
    int K, int N,
    const float* __restrict__ bias,   // EPI0/1: bias; EPI2: rn; EPI3: rowsum
    const float* __restrict__ aux0,   // EPI0: te|null; EPI1: x; EPI3: xn
    const float* __restrict__ aux1,   // EPI1: glp;     EPI3: xold
    const float* __restrict__ aux2,   // EPI1: noise;   EPI3: scal
    const float* __restrict__ epsp,
    float* __restrict__ Out)
{
    __shared__ float lb[16 * 128];        // B tile, pair-interleaved (32k x 64n)
#if HAVE_ASYNC_LDS
    __shared__ float lat[64 * ASTR];      // A tile, padded rows (64m x 32k)
#endif

    const int tid   = threadIdx.x;
    const int wv    = tid >> 5;
    const int ln    = tid & 31;
    const int l15   = ln & 15;
    const int khalf = (ln < 16) ? 0 : 2;
    const int baseN = blockIdx.x * 64;
    const int baseM = blockIdx.y * 64;
    const int rowA  = baseM + wv * 16 + l15;
    const int rowL  = wv * 16 + l15;
    (void)rowA; (void)rowL;

    v8f acc[4];
#pragma unroll
    for (int i = 0; i < 4; ++i)
#pragma unroll
        for (int j = 0; j < 8; ++j) acc[i][j] = 0.0f;

    for (int k0 = 0; k0 < K; k0 += 32) {
#if HAVE_ASYNC_LDS
        // ---- A tile: async DMA global -> LDS (4 x 16B per lane) ----
        {
            typedef __attribute__((address_space(1))) i32x4* gp_t;
            typedef __attribute__((address_space(3))) i32x4* lp_t;
#pragma unroll
            for (int j = 0; j < 4; ++j) {
                const int q  = tid + 128 * j;
                const int r  = q >> 3, c8 = q & 7;
                __builtin_amdgcn_global_load_async_to_lds_b128(
                    (gp_t)(A + (size_t)(baseM + r) * K + k0 + c8 * 4),
                    (lp_t)(lat + r * ASTR + c8 * 4), 0, 0);
            }
        }
#endif
        if (!BT) {
            if (k0 + 32 < K) {
                // prefetch next chunk's B rows into cache
                __builtin_prefetch(Bm + (size_t)(k0 + 32 + (tid >> 2)) * N + baseN + (tid & 3) * 16, 0, 3);
            }
            // ---- B tile: load 4 float4 (2 row-pairs), interleave, store 2x b128 ----
            float4 g0[2], g1[2];
#pragma unroll
            for (int r = 0; r < 2; ++r) {
                const int q  = tid + 128 * r;
                const int kp = q >> 4, n4 = q & 15;
                g0[r] = *(const float4*)(Bm + (size_t)(k0 + 2 * kp) * N + baseN + n4 * 4);
                g1[r] = *(const float4*)(Bm + (size_t)(k0 + 2 * kp + 1) * N + baseN + n4 * 4);
            }
#pragma unroll
            for (int r = 0; r < 2; ++r) {
                const int q  = tid + 128 * r;
                const int kp = q >> 4, n4 = q & 15;
                float* dst = lb + kp * 128 + n4 * 8;
                *(float4*)(dst)     = make_float4(g0[r].x, g1[r].x, g0[r].y, g1[r].y);
                *(float4*)(dst + 4) = make_float4(g0[r].z, g1[r].z, g0[r].w, g1[r].w);
            }
        } else {
            // B[k][n] = Bm[n][k] (Gram X @ X^T): 4 float4 along k per thread
            float4 gb[4];
#pragma unroll
            for (int r = 0; r < 4; ++r) {
                const int q  = tid + 128 * r;
                const int k4 = q >> 6, nn = q & 63;
                gb[r] = *(const float4*)(Bm + (size_t)(baseN + nn) * K + k0 + k4 * 4);
            }
#pragma unroll
            for (int r = 0; r < 4; ++r) {
                const int q  = tid + 128 * r;
                const int k4 = q >> 6, nn = q & 63;
                *(float2*)(lb + (2 * k4) * 128 + nn * 2)     = make_float2(gb[r].x, gb[r].y);
                *(float2*)(lb + (2 * k4 + 1) * 128 + nn * 2) = make_float2(gb[r].z, gb[r].w);
            }
        }
        wait_async_zero();
        __syncthreads();

#pragma unroll
        for (int kk = 0; kk < 32; kk += 4) {
#if HAVE_ASYNC_LDS
            const v2f a = *(const v2f*)(lat + rowL * ASTR + kk + khalf);
#else
            const v2f a = *(const v2f*)(A + (size_t)rowA * K + k0 + kk + khalf);
#endif
            const int   kp = (kk + khalf) >> 1;
            const float* bp = lb + kp * 128 + l15 * 2;
#pragma unroll
            for (int nt = 0; nt < 4; ++nt) {
                const v2f b = *(const v2f*)(bp + nt * 32);
                acc[nt] = __builtin_amdgcn_wmma_f32_16x16x4_f32(
                    false, a, false, b, (short)0, acc[nt], false, false);
            }
        }
        __syncthreads();
    }

    float dt = 0.0f, s2 = 0.0f, coef = 0.0f;
    if (EPI == 1) { dt = epsp[0]; s2 = sqrtf(2.0f * dt); }
    if (EPI == 3) { coef = aux2[1]; }

#pragma unroll
    for (int nt = 0; nt < 4; ++nt) {
        const int col = baseN + nt * 16 + l15;
#pragma unroll
        for (int v = 0; v < 8; ++v) {
            const int row = baseM + wv * 16 + v + ((ln < 16) ? 0 : 8);
            const float val = acc[nt][v];
            const size_t oidx = (size_t)row * N + col;
            if (EPI == 0) {
                float s = val + bias[col];
                if (aux0) s += aux0[col];
                Out[oidx] = gelu_f(s);
            } else if (EPI == 1) {
                const float y = val + bias[col];
                Out[oidx] = aux0[oidx] + dt * (aux1[oidx] - y) + s2 * aux2[oidx];
            } else if (EPI == 2) {
                Out[oidx] = bias[row] + bias[col] - 2.0f * val;
            } else {
                Out[oidx] = aux0[oidx] + coef * (bias[row] * aux1[oidx] - val);
            }
        }
    }
}

// ---------------------------------------------------------------------------
// Per-step prep: betas (block 0) + time embeddings te[i] (one block per step)
// ---------------------------------------------------------------------------
__global__ __launch_bounds__(256) void prep_k(
    const float* __restrict__ grid_t, const float* __restrict__ phase,
    const float* __restrict__ tW1, const float* __restrict__ tb1,
    const float* __restrict__ tW2, const float* __restrict__ tb2,
    float* __restrict__ te, float* __restrict__ betas)
{
    __shared__ float temb[2 * Cc];
    __shared__ float g1[Cc];
    const int i = blockIdx.x;
    const float t = (float)i;

    for (int c = threadIdx.x; c < Cc; c += 256) {
        float e = (0.1f + c * (99.9f / 511.0f)) * t + phase[c];
        temb[c]      = sinf(e);
        temb[Cc + c] = cosf(e);
    }
    if (i == 0 && threadIdx.x == 0) {
        float s[NB]; float tot = 0.0f;
        for (int j = 0; j < NB; ++j) { s[j] = 1.0f / (1.0f + expf(-grid_t[j])); tot += s[j]; }
        float acc = 0.0f; betas[0] = 0.0f;
        for (int j = 0; j < NB; ++j) { acc += s[j]; betas[j + 1] = acc / tot; }
    }
    __syncthreads();
    for (int c = threadIdx.x; c < Cc; c += 256) {
        float s = tb1[c];
        for (int k = 0; k < 2 * Cc; ++k) s += temb[k] * tW1[(size_t)k * Cc + c];
        g1[c] = gelu_f(s);
    }
    __syncthreads();
    for (int c = threadIdx.x; c < Cc; c += 256) {
        float s = tb2[c];
        for (int k = 0; k < Cc; ++k) s += g1[k] * tW2[(size_t)k * Cc + c];
        te[(size_t)i * Cc + c] = s;
    }
}

// ---------------------------------------------------------------------------
// grad log pi (annealed) + row norms. One 64-thread block per particle.
// ---------------------------------------------------------------------------
__global__ __launch_bounds__(64) void glp_rn_k(
    const float* __restrict__ x, const float* __restrict__ betas, int step,
    const float* __restrict__ means, float* __restrict__ glp, float* __restrict__ rn)
{
    __shared__ float red[64];
    __shared__ float comp[8];
    __shared__ float wgt[8];
    const int row = blockIdx.x, d = threadIdx.x;
    const float xd = x[(size_t)row * Dd + d];

    red[d] = xd * xd; __syncthreads();
    for (int s = 32; s > 0; s >>= 1) { if (d < s) red[d] += red[d + s]; __syncthreads(); }
    if (d == 0) rn[row] = red[0];

    const float t = betas[step];
    for (int m = 0; m < 8; ++m) {
        __syncthreads();
        float diff = xd - means[(size_t)m * Dd + d];
        red[d] = diff * diff; __syncthreads();
        for (int s = 32; s > 0; s >>= 1) { if (d < s) red[d] += red[d + s]; __syncthreads(); }
        if (d == 0) comp[m] = -0.5f * (red[0] + 64.0f * LOG2PI);
    }
    __syncthreads();
    if (d == 0) {
        float mx = comp[0];
        for (int m = 1; m < 8; ++m) mx = fmaxf(mx, comp[m]);
        float sum = 0.0f;
        for (int m = 0; m < 8; ++m) { wgt[m] = expf(comp[m] - mx); sum += wgt[m]; }
        for (int m = 0; m < 8; ++m) wgt[m] /= sum;
    }
    __syncthreads();
    float acc = (1.0f - t) * (-xd);
    for (int m = 0; m < 8; ++m) acc += t * wgt[m] * (means[(size_t)m * Dd + d] - xd);
    glp[(size_t)row * Dd + d] = acc;
}

// ---------------------------------------------------------------------------
// Median of dist = sqrt(where(d2>0,d2,1e-12)) over 512*512 values.
// 31-pass bit-bisection for rank N/2, then ONE pass for rank N/2+1:
// count(<= m1) decides duplicate vs min-of-greater (atomicMin on bits).
// Writes scal[0] = 1/h_t, scal[1] = 0.1*dt/h_t.
// ---------------------------------------------------------------------------
__global__ __launch_bounds__(1024) void median_k(
    const float* __restrict__ d2, const float* __restrict__ epsp, float* __restrict__ scal)
{
    __shared__ unsigned s_cnt, s_lo, s_hi, s_min;
    const int n = Bn * Bn;
    const unsigned k = (unsigned)(n / 2);       // 1-based rank of lower middle

    if (threadIdx.x == 0) { s_lo = 0u; s_hi = 0x7F800000u; }
    __syncthreads();
    for (int it = 0; it < 31; ++it) {
        const unsigned lo = s_lo, hi = s_hi;
        const unsigned mid = lo + ((hi - lo) >> 1);
        if (threadIdx.x == 0) s_cnt = 0u;
        __syncthreads();
        unsigned c = 0;
        for (int idx = threadIdx.x; idx < n; idx += 1024) {
            float v = d2[idx];
            v = (v > 0.0f) ? v : 1e-12f;
            if (__float_as_uint(v) <= mid) ++c;
        }
        atomicAdd(&s_cnt, c);
        __syncthreads();
        if (threadIdx.x == 0) { if (s_cnt >= k) s_hi = mid; else s_lo = mid + 1u; }
        __syncthreads();
    }
    const unsigned m1 = s_hi;
    if (threadIdx.x == 0) { s_cnt = 0u; s_min = 0x7F800000u; }
    __syncthreads();
    {
        unsigned c = 0, mn = 0x7F800000u;
        for (int idx = threadIdx.x; idx < n; idx += 1024) {
            float v = d2[idx];
            v = (v > 0.0f) ? v : 1e-12f;
            const unsigned u = __float_as_uint(v);
            if (u <= m1) ++c; else mn = (u < mn) ? u : mn;
        }
        atomicAdd(&s_cnt, c);
        atomicMin(&s_min, mn);
    }
    __syncthreads();
    if (threadIdx.x == 0) {
        const float v1 = __uint_as_float(m1);
        const float v2 = (s_cnt >= k + 1u) ? v1 : __uint_as_float(s_min);
        const float med = 0.5f * (sqrtf(v1) + sqrtf(v2));
        const float ht  = med * med / logf(512.0f);
        const float dt  = epsp[0];
        scal[0] = 1.0f / ht;
        scal[1] = 0.1f * dt / ht;
    }
}

// ---------------------------------------------------------------------------
// W = exp(-d2/h_t) (in place) + row sums. One 128-thread block per row.
// ---------------------------------------------------------------------------
__global__ __launch_bounds__(128) void wmat_k(
    float* __restrict__ d2, const float* __restrict__ scal, float* __restrict__ rowsum)
{
    __shared__ float red[128];
    const int row = blockIdx.x;
    const float inv_ht = scal[0];
    float s = 0.0f;
    for (int j = threadIdx.x; j < Bn; j += 128) {
        const size_t idx = (size_t)row * Bn + j;
        const float wv = expf(-d2[idx] * inv_ht);
        d2[idx] = wv;
        s += wv;
    }
    red[threadIdx.x] = s; __syncthreads();
    for (int k = 64; k > 0; k >>= 1) { if (threadIdx.x < k) red[threadIdx.x] += red[threadIdx.x + k]; __syncthreads(); }
    if (threadIdx.x == 0) rowsum[row] = red[0];
}

__global__ void copy_k(const float* __restrict__ src, float* __restrict__ dst, int n)
{
    int i = blockIdx.x * blockDim.x + threadIdx.x;
    if (i < n) dst[i] = src[i];
}

// ---------------------------------------------------------------------------
extern "C" void kernel_launch(void* const* d_in, const int* in_sizes, int n_in,
                              void* d_out, int out_size, void* d_ws, size_t ws_size,
                              hipStream_t stream)
{
    (void)in_sizes; (void)n_in; (void)out_size; (void)ws_size;

    const float* particles = (const float*)d_in[0];
    const float* noises    = (const float*)d_in[1];
    const float* grid_t    = (const float*)d_in[2];
    const float* eps       = (const float*)d_in[3];
    const float* means     = (const float*)d_in[4];
    const float* phase     = (const float*)d_in[5];
    const float* in_W      = (const float*)d_in[6];
    const float* in_b      = (const float*)d_in[7];
    const float* tW1       = (const float*)d_in[8];
    const float* tb1       = (const float*)d_in[9];
    const float* tW2       = (const float*)d_in[10];
    const float* tb2       = (const float*)d_in[11];
    const float* hW        = (const float*)d_in[12];
    const float* hb        = (const float*)d_in[13];
    const float* outW      = (const float*)d_in[14];
    const float* outb      = (const float*)d_in[15];

    float* out = (float*)d_out;
    float* w   = (float*)d_ws;

    const size_t BD = (size_t)Bn * Dd;          // 32768
    const size_t CC = (size_t)Cc * Cc;          // 262144
    float* betas  = w;                  // 16
    float* te     = betas + 16;         // 8*512
    float* glp    = te + 8 * Cc;        // 32768
    float* rn     = glp + BD;           // 512
    float* scal   = rn + Bn;            // 16
    float* rowsum = scal + 16;          // 512
    float* h1     = rowsum + Bn;        // 262144
    float* h2     = h1 + CC;            // 262144
    float* xn     = h2 + CC;            // 32768
    float* d2     = xn + BD;            // 262144

    copy_k<<<(int)((BD + 255) / 256), 256, 0, stream>>>(particles, out, (int)BD);
    prep_k<<<NB, 256, 0, stream>>>(grid_t, phase, tW1, tb1, tW2, tb2, te, betas);

    for (int i = 0; i < NB; ++i) {
        const float* x  = out + (size_t)i * BD;
        float*       xo = out + (size_t)(i + 1) * BD;
        const float* nz = noises + (size_t)i * BD;

        glp_rn_k<<<Bn, 64, 0, stream>>>(x, betas, i, means, glp, rn);

        // h1 = gelu(x @ in_W + in_b + te_i)
        gemm_k<0, false><<<dim3(8, 8), 128, 0, stream>>>(x, in_W, Dd, Cc, in_b, te + (size_t)i * Cc, nullptr, nullptr, eps, h1);
        // three hidden layers
        gemm_k<0, false><<<dim3(8, 8), 128, 0, stream>>>(h1, hW + 0 * CC, Cc, Cc, hb + 0 * Cc, nullptr, nullptr, nullptr, eps, h2);
        gemm_k<0, false><<<dim3(8, 8), 128, 0, stream>>>(h2, hW + 1 * CC, Cc, Cc, hb + 1 * Cc, nullptr, nullptr, nullptr, eps, h1);
        gemm_k<0, false><<<dim3(8, 8), 128, 0, stream>>>(h1, hW + 2 * CC, Cc, Cc, hb + 2 * Cc, nullptr, nullptr, nullptr, eps, h2);
        // xn = x + dt*(glp - (h2 @ out_W + out_b)) + sqrt(2dt)*noise
        gemm_k<1, false><<<dim3(1, 8), 128, 0, stream>>>(h2, outW, Cc, Dd, outb, x, glp, nz, eps, xn);
        // d2 = rn_i + rn_j - 2 * (X X^T)
        gemm_k<2, true ><<<dim3(8, 8), 128, 0, stream>>>(x, x, Dd, Bn, rn, nullptr, nullptr, nullptr, eps, d2);
        // median-heuristic bandwidth -> scal = {1/h_t, 0.1*dt/h_t}
        median_k<<<1, 1024, 0, stream>>>(d2, eps, scal);
        // W = exp(-d2/h_t) in place, rowsum
        wmat_k<<<Bn, 128, 0, stream>>>(d2, scal, rowsum);
        // xo = xn + coef*(rowsum*x - W @ x)
        gemm_k<3, false><<<dim3(1, 8), 128, 0, stream>>>(d2, x, Bn, Dd, rowsum, xn, x, scal, eps, xo);
    }
}